// Matrix_58875411693702
// MI455X (gfx1250) — compile-verified
//
#include <hip/hip_runtime.h>
#include <math.h>

typedef __attribute__((ext_vector_type(2))) float v2f;
typedef __attribute__((ext_vector_type(8))) float v8f;

#define K_SPK 160
#define T_DIM 512
#define D_DIM 40
#define L_DIM (T_DIM * D_DIM) /* 20480 */
#define EPS_N 1e-12f

__device__ __forceinline__ float softplus_f(float x) {
    // numerically stable softplus
    return fmaxf(x, 0.0f) + log1pf(expf(-fabsf(x)));
}

// ---------------------------------------------------------------------------
// Kernel 1: L2-normalize enroll over the T axis (per (k,d) column) -> Abuf
// Block = one speaker k. 320 threads = 8 t-rows x 40 d-lanes (coalesced in d).
// ---------------------------------------------------------------------------
__global__ __launch_bounds__(320) void norm_enroll_kernel(const float* __restrict__ enroll,
                                                          float* __restrict__ Abuf) {
    __shared__ float ssbuf[320];
    __shared__ float inv[D_DIM];
    const int k   = blockIdx.x;
    const int tid = threadIdx.x;
    const int d   = tid % D_DIM;
    const int tr  = tid / D_DIM; // 0..7

    const float* src = enroll + (size_t)k * L_DIM;
    float ss = 0.f;
    for (int t = tr; t < T_DIM; t += 8) {
        float x = src[t * D_DIM + d];
        ss = fmaf(x, x, ss);
    }
    ssbuf[tid] = ss;
    __syncthreads();
    if (tid < D_DIM) {
        float tot = 0.f;
        #pragma unroll
        for (int r = 0; r < 8; ++r) tot += ssbuf[r * D_DIM + tid];
        inv[tid] = 1.0f / fmaxf(sqrtf(tot), EPS_N);
    }
    __syncthreads();
    const float iv = inv[d];
    float* dst = Abuf + (size_t)k * L_DIM;
    for (int t = tr; t < T_DIM; t += 8) {
        dst[t * D_DIM + d] = src[t * D_DIM + d] * iv;
    }
}

// ---------------------------------------------------------------------------
// Kernel 2: L2-normalize test over T, then U[k,t,d] = sum_e softplus(W)[d,e] *
// t_hat[k,t,e]  -> Bbuf.  softplus(W) staged in LDS; matvec done on 8 rows at
// a time staged (already normalized) in LDS.
// ---------------------------------------------------------------------------
__global__ __launch_bounds__(320) void norm_test_transform_kernel(const float* __restrict__ test,
                                                                  const float* __restrict__ weight,
                                                                  float* __restrict__ Bbuf) {
    __shared__ float sp[D_DIM * D_DIM]; // softplus(W), 6.4 KB
    __shared__ float ssbuf[320];
    __shared__ float inv[D_DIM];
    __shared__ float rowbuf[8 * D_DIM];

    const int k   = blockIdx.x;
    const int tid = threadIdx.x;
    const int d   = tid % D_DIM;
    const int tr  = tid / D_DIM;

    for (int i = tid; i < D_DIM * D_DIM; i += 320) sp[i] = softplus_f(weight[i]);

    const float* src = test + (size_t)k * L_DIM;
    float ss = 0.f;
    for (int t = tr; t < T_DIM; t += 8) {
        float x = src[t * D_DIM + d];
        ss = fmaf(x, x, ss);
    }
    ssbuf[tid] = ss;
    __syncthreads();
    if (tid < D_DIM) {
        float tot = 0.f;
        #pragma unroll
        for (int r = 0; r < 8; ++r) tot += ssbuf[r * D_DIM + tid];
        inv[tid] = 1.0f / fmaxf(sqrtf(tot), EPS_N);
    }
    __syncthreads();
    const float iv = inv[d];
    float* dst = Bbuf + (size_t)k * L_DIM;

    for (int t0 = 0; t0 < T_DIM; t0 += 8) {
        const int t = t0 + tr;
        rowbuf[tr * D_DIM + d] = src[t * D_DIM + d] * iv; // normalized row element
        __syncthreads();
        float u = 0.f;
        const float* spr = &sp[d * D_DIM];
        const float* rb  = &rowbuf[tr * D_DIM];
        #pragma unroll
        for (int e = 0; e < D_DIM; ++e) u = fmaf(spr[e], rb[e], u);
        dst[t * D_DIM + d] = u;
        __syncthreads();
    }
}

// ---------------------------------------------------------------------------
// Kernel 3: out = A * B^T, M=N=160, K=20480, fp32 WMMA 16x16x4.
// Block = one 32x32 output macro-tile (grid 5x5); each of the 8 waves keeps a
// 2x2 group of 16x16 accumulator tiles (register reuse: 4 loads feed 4 WMMAs)
// over a 2560-long K slice.  Wave id is readfirstlane'd so the K-loop bounds
// live in SGPRs (scalar loop control, no v_cmpx/EXEC predication around the
// WMMAs).  Explicit double-buffered software pipeline: all 16 loads of the
// NEXT 16-wide K chunk go into a separate register buffer before the 16 WMMAs
// of the CURRENT chunk execute.  Deterministic fixed-order accumulation;
// 8 partial 32x32 tiles summed through 32 KB of LDS.
// ---------------------------------------------------------------------------
__global__ __launch_bounds__(256) void gemm_wmma_kernel(const float* __restrict__ A,
                                                        const float* __restrict__ B,
                                                        float* __restrict__ out) {
    __shared__ float cred[8 * 1024];
    const int i0   = blockIdx.x * 32;
    const int j0   = blockIdx.y * 32;
    const int tid  = threadIdx.x;
    // wave id is uniform across the wave: pin it (and the loop bounds) to SGPRs
    const int wave = __builtin_amdgcn_readfirstlane(tid >> 5); // 0..7
    const int lane = tid & 31;
    const int r    = lane & 15;
    const int h    = lane >> 4;

    const float* pA0 = A + (size_t)(i0 + r)      * L_DIM + 2 * h;
    const float* pA1 = A + (size_t)(i0 + 16 + r) * L_DIM + 2 * h;
    const float* pB0 = B + (size_t)(j0 + r)      * L_DIM + 2 * h;
    const float* pB1 = B + (size_t)(j0 + 16 + r) * L_DIM + 2 * h;
    const int kbeg = wave * (L_DIM / 8);
    const int kend = kbeg + (L_DIM / 8); // slice = 2560 = 160 chunks of 16

    v8f c00 = {}, c01 = {}, c10 = {}, c11 = {};

    // Two operand register buffers (x and y), 16 loads each.
    v2f xa0[4], xa1[4], xb0[4], xb1[4];
    v2f ya0[4], ya1[4], yb0[4], yb1[4];

    auto loadx = [&](int kk) {
        #pragma unroll
        for (int s = 0; s < 4; ++s) {
            xa0[s] = *(const v2f*)(pA0 + kk + 4 * s);
            xa1[s] = *(const v2f*)(pA1 + kk + 4 * s);
            xb0[s] = *(const v2f*)(pB0 + kk + 4 * s);
            xb1[s] = *(const v2f*)(pB1 + kk + 4 * s);
        }
    };
    auto loady = [&](int kk) {
        #pragma unroll
        for (int s = 0; s < 4; ++s) {
            ya0[s] = *(const v2f*)(pA0 + kk + 4 * s);
            ya1[s] = *(const v2f*)(pA1 + kk + 4 * s);
            yb0[s] = *(const v2f*)(pB0 + kk + 4 * s);
            yb1[s] = *(const v2f*)(pB1 + kk + 4 * s);
        }
    };
    auto compx = [&]() {
        #pragma unroll
        for (int s = 0; s < 4; ++s) {
            c00 = __builtin_amdgcn_wmma_f32_16x16x4_f32(false, xa0[s], false, xb0[s], (short)0, c00, false, false);
            c01 = __builtin_amdgcn_wmma_f32_16x16x4_f32(false, xa0[s], false, xb1[s], (short)0, c01, false, false);
            c10 = __builtin_amdgcn_wmma_f32_16x16x4_f32(false, xa1[s], false, xb0[s], (short)0, c10, false, false);
            c11 = __builtin_amdgcn_wmma_f32_16x16x4_f32(false, xa1[s], false, xb1[s], (short)0, c11, false, false);
        }
    };
    auto compy = [&]() {
        #pragma unroll
        for (int s = 0; s < 4; ++s) {
            c00 = __builtin_amdgcn_wmma_f32_16x16x4_f32(false, ya0[s], false, yb0[s], (short)0, c00, false, false);
            c01 = __builtin_amdgcn_wmma_f32_16x16x4_f32(false, ya0[s], false, yb1[s], (short)0, c01, false, false);
            c10 = __builtin_amdgcn_wmma_f32_16x16x4_f32(false, ya1[s], false, yb0[s], (short)0, c10, false, false);
            c11 = __builtin_amdgcn_wmma_f32_16x16x4_f32(false, ya1[s], false, yb1[s], (short)0, c11, false, false);
        }
    };

    // Software pipeline: two 16-float chunks (x and y) per iteration.
    loadx(kbeg);
    int kk = kbeg;
    #pragma clang loop unroll(disable)
    for (; kk < kend - 32; kk += 32) {
        loady(kk + 16);  // prefetch chunk n+1 while computing chunk n
        compx();
        loadx(kk + 32);  // prefetch chunk n+2 while computing chunk n+1
        compy();
    }
    // Epilogue: kk == kend - 32; x holds chunk at kk, load+compute final chunk.
    loady(kk + 16);
    compx();
    compy();

    float* red = &cred[wave * 1024];
    #pragma unroll
    for (int v = 0; v < 8; ++v) {
        red[0 * 256 + v * 32 + lane] = c00[v]; // rowTile 0, colTile 0
        red[1 * 256 + v * 32 + lane] = c01[v]; // rowTile 0, colTile 1
        red[2 * 256 + v * 32 + lane] = c10[v]; // rowTile 1, colTile 0
        red[3 * 256 + v * 32 + lane] = c11[v]; // rowTile 1, colTile 1
    }
    __syncthreads();

    // Sum the 8 per-wave partial 32x32 tiles; 256 threads x 4 elems each.
    #pragma unroll
    for (int e = 0; e < 4; ++e) {
        const int idx = e * 256 + tid; // 0..1023
        float s = 0.f;
        #pragma unroll
        for (int w = 0; w < 8; ++w) s += cred[w * 1024 + idx];
        const int ti  = idx >> 8;     // tile id 0..3
        const int rem = idx & 255;
        const int v   = rem >> 5;     // accumulator VGPR index
        const int l   = rem & 31;
        const int hh  = l >> 4;
        const int n   = l & 15;
        const int row = i0 + ((ti >> 1) << 4) + v + 8 * hh; // M = v + 8*(lane/16)
        const int col = j0 + ((ti & 1) << 4) + n;           // N = lane%16
        out[row * K_SPK + col] = s;
    }
}

extern "C" void kernel_launch(void* const* d_in, const int* in_sizes, int n_in,
                              void* d_out, int out_size, void* d_ws, size_t ws_size,
                              hipStream_t stream) {
    (void)in_sizes; (void)n_in; (void)out_size; (void)ws_size;
    const float* enroll = (const float*)d_in[0]; // [160,512,40]
    const float* test   = (const float*)d_in[1]; // [160,512,40]
    const float* weight = (const float*)d_in[2]; // [40,40]
    float* out  = (float*)d_out;                 // [160,160]
    float* Abuf = (float*)d_ws;                  // [160,512,40] normalized enroll
    float* Bbuf = Abuf + (size_t)K_SPK * L_DIM;  // [160,512,40] W+ @ normalized test

    norm_enroll_kernel<<<K_SPK, 320, 0, stream>>>(enroll, Abuf);
    norm_test_transform_kernel<<<K_SPK, 320, 0, stream>>>(test, weight, Bbuf);
    gemm_wmma_kernel<<<dim3(5, 5), 256, 0, stream>>>(Abuf, Bbuf, out);
}